// LiftedStructureLoss_45028437131621
// MI455X (gfx1250) — compile-verified
//
#include <hip/hip_runtime.h>
#include <hip/hip_bf16.h>
#include <math.h>

#define NROW 8192
#define DIM  128
#define NTILE (NROW / 16)   // 512 tiles per dimension
#define BASE_M 0.5f
#define HARD_EPS 0.1f

typedef float v2f __attribute__((ext_vector_type(2)));
typedef float v8f __attribute__((ext_vector_type(8)));

__device__ __forceinline__ v8f wmma_f32_k4(v2f a, v2f b, v8f c) {
  return __builtin_amdgcn_wmma_f32_16x16x4_f32(
      /*neg_a=*/false, a, /*neg_b=*/false, b,
      /*c_mod=*/(short)0, c, /*reuse_a=*/false, /*reuse_b=*/false);
}

// Compute one 16x16 sim tile with two independent WMMA accumulator chains
// (k = 0..15 and k = 16..31) to break the serial accumulator dependency.
__device__ __forceinline__ v8f sim_tile(const v2f* __restrict__ afrag,
                                        const float* __restrict__ bp) {
  v8f acc0 = {}, acc1 = {};
  #pragma unroll
  for (int k = 0; k < 16; ++k) {
    v2f b0 = *(const v2f*)(bp + 4 * k);
    v2f b1 = *(const v2f*)(bp + 4 * (k + 16));
    acc0 = wmma_f32_k4(afrag[k],      b0, acc0);
    acc1 = wmma_f32_k4(afrag[k + 16], b1, acc1);
  }
  return acc0 + acc1;
}

// ---------------------------------------------------------------------------
// Pass 1: per-row min positive sim, max negative sim, and raw pos/neg sums.
// One block = one 16-row tile; 8 waves stride over the 512 column tiles.
// ---------------------------------------------------------------------------
__global__ __launch_bounds__(256) void lifted_pass1(
    const float* __restrict__ X, const int* __restrict__ T,
    float* __restrict__ minpos, float* __restrict__ maxneg,
    float* __restrict__ posd,   float* __restrict__ negd)
{
  const int itile  = blockIdx.x;           // row tile 0..511
  const int wave   = threadIdx.x >> 5;     // 0..7
  const int lane   = threadIdx.x & 31;
  const int lane16 = lane & 15;
  const int half   = lane >> 4;            // 0: k0..k0+1, rows 0..7 of C; 1: k0+2..3, rows 8..15
  const int rowbase = itile * 16;

  // Preload all 32 A fragments (float2 per k-step) for this lane's A row.
  v2f afrag[32];
  {
    const float* ap = X + (rowbase + lane16) * DIM + half * 2;
    #pragma unroll
    for (int k = 0; k < 32; ++k)
      afrag[k] = *(const v2f*)(ap + 4 * k);
  }
  // Targets of the 8 C rows this lane accumulates.
  int trow[8];
  #pragma unroll
  for (int v = 0; v < 8; ++v) trow[v] = T[rowbase + half * 8 + v];

  float mx[8], mn[8], ps[8], ns[8];
  #pragma unroll
  for (int v = 0; v < 8; ++v) { mx[v] = -INFINITY; mn[v] = INFINITY; ps[v] = 0.f; ns[v] = 0.f; }

  for (int jt = wave; jt < NTILE; jt += 8) {
    const int colbase = jt * 16;
    const float* bp = X + (colbase + lane16) * DIM + half * 2;
    const int tcol = T[colbase + lane16];

    v8f acc = sim_tile(afrag, bp);

    const int col = colbase + lane16;
    #pragma unroll
    for (int v = 0; v < 8; ++v) {
      const int row = rowbase + half * 8 + v;
      const float s = acc[v];
      const bool same = (trow[v] == tcol);
      if (same) {
        if (row != col) { mn[v] = fminf(mn[v], s); ps[v] += s; }
      } else {
        mx[v] = fmaxf(mx[v], s); ns[v] += s;
      }
    }
  }

  // Reduce across the 16 lanes of each half (halves stay independent).
  #pragma unroll
  for (int off = 8; off >= 1; off >>= 1) {
    #pragma unroll
    for (int v = 0; v < 8; ++v) {
      mx[v] = fmaxf(mx[v], __shfl_xor(mx[v], off, 16));
      mn[v] = fminf(mn[v], __shfl_xor(mn[v], off, 16));
      ps[v] += __shfl_xor(ps[v], off, 16);
      ns[v] += __shfl_xor(ns[v], off, 16);
    }
  }

  __shared__ float smx[8][16], smn[8][16], sps[8][16], sns[8][16];
  if (lane16 == 0) {
    #pragma unroll
    for (int v = 0; v < 8; ++v) {
      const int r = half * 8 + v;
      smx[wave][r] = mx[v]; smn[wave][r] = mn[v];
      sps[wave][r] = ps[v]; sns[wave][r] = ns[v];
    }
  }
  __syncthreads();

  if (threadIdx.x < 16) {
    const int r = threadIdx.x;
    float m = -INFINITY, n2 = INFINITY, p = 0.f, q = 0.f;
    #pragma unroll
    for (int w = 0; w < 8; ++w) {
      m = fmaxf(m, smx[w][r]); n2 = fminf(n2, smn[w][r]);
      p += sps[w][r];          q += sns[w][r];
    }
    maxneg[rowbase + r] = m;  minpos[rowbase + r] = n2;
    posd[rowbase + r]   = p;  negd[rowbase + r]   = q;
  }
}

// ---------------------------------------------------------------------------
// Pass 2: recompute sim tiles, apply hard-mining validity masks, accumulate
// masked exp sums and valid counts per row.
// ---------------------------------------------------------------------------
__global__ __launch_bounds__(256) void lifted_pass2(
    const float* __restrict__ X, const int* __restrict__ T,
    const float* __restrict__ minpos, const float* __restrict__ maxneg,
    float* __restrict__ pexp, float* __restrict__ nexp,
    float* __restrict__ pcnt, float* __restrict__ ncnt)
{
  const int itile  = blockIdx.x;
  const int wave   = threadIdx.x >> 5;
  const int lane   = threadIdx.x & 31;
  const int lane16 = lane & 15;
  const int half   = lane >> 4;
  const int rowbase = itile * 16;

  v2f afrag[32];
  {
    const float* ap = X + (rowbase + lane16) * DIM + half * 2;
    #pragma unroll
    for (int k = 0; k < 32; ++k)
      afrag[k] = *(const v2f*)(ap + 4 * k);
  }
  int trow[8];
  float mnp[8], mxn[8];
  #pragma unroll
  for (int v = 0; v < 8; ++v) {
    const int r = rowbase + half * 8 + v;
    trow[v] = T[r]; mnp[v] = minpos[r]; mxn[v] = maxneg[r];
  }

  float pe[8], ne[8], pc[8], nc[8];
  #pragma unroll
  for (int v = 0; v < 8; ++v) { pe[v] = 0.f; ne[v] = 0.f; pc[v] = 0.f; nc[v] = 0.f; }

  for (int jt = wave; jt < NTILE; jt += 8) {
    const int colbase = jt * 16;
    const float* bp = X + (colbase + lane16) * DIM + half * 2;
    const int tcol = T[colbase + lane16];

    v8f acc = sim_tile(afrag, bp);

    const int col = colbase + lane16;
    #pragma unroll
    for (int v = 0; v < 8; ++v) {
      const int row = rowbase + half * 8 + v;
      const float s = acc[v];
      const bool same = (trow[v] == tcol);
      if (same) {
        if (row != col && (s - HARD_EPS < mxn[v])) {
          pe[v] += __expf(-(s - BASE_M)); pc[v] += 1.f;
        }
      } else if (s + HARD_EPS > mnp[v]) {
        ne[v] += __expf(s - BASE_M); nc[v] += 1.f;
      }
    }
  }

  #pragma unroll
  for (int off = 8; off >= 1; off >>= 1) {
    #pragma unroll
    for (int v = 0; v < 8; ++v) {
      pe[v] += __shfl_xor(pe[v], off, 16);
      ne[v] += __shfl_xor(ne[v], off, 16);
      pc[v] += __shfl_xor(pc[v], off, 16);
      nc[v] += __shfl_xor(nc[v], off, 16);
    }
  }

  __shared__ float spe[8][16], sne[8][16], spc[8][16], snc[8][16];
  if (lane16 == 0) {
    #pragma unroll
    for (int v = 0; v < 8; ++v) {
      const int r = half * 8 + v;
      spe[wave][r] = pe[v]; sne[wave][r] = ne[v];
      spc[wave][r] = pc[v]; snc[wave][r] = nc[v];
    }
  }
  __syncthreads();

  if (threadIdx.x < 16) {
    const int r = threadIdx.x;
    float a = 0.f, b = 0.f, c = 0.f, d = 0.f;
    #pragma unroll
    for (int w = 0; w < 8; ++w) {
      a += spe[w][r]; b += sne[w][r]; c += spc[w][r]; d += snc[w][r];
    }
    pexp[rowbase + r] = a; nexp[rowbase + r] = b;
    pcnt[rowbase + r] = c; ncnt[rowbase + r] = d;
  }
}

// ---------------------------------------------------------------------------
// Pass 3: fold 8192 rows into the 4 scalar outputs.
// ---------------------------------------------------------------------------
__global__ __launch_bounds__(256) void lifted_finalize(
    const float* __restrict__ pexp, const float* __restrict__ nexp,
    const float* __restrict__ pcnt, const float* __restrict__ ncnt,
    const float* __restrict__ posd, const float* __restrict__ negd,
    float* __restrict__ out)
{
  const int tid = threadIdx.x;
  float loss = 0.f, skip = 0.f, pd = 0.f, nd = 0.f;
  for (int r = tid; r < NROW; r += 256) {
    const bool has = (pcnt[r] > 0.f) && (ncnt[r] > 0.f);
    if (has) loss += logf(pexp[r]) + logf(nexp[r]);
    else     skip += 1.f;
    pd += posd[r]; nd += negd[r];
  }
  __shared__ float red[4][256];
  red[0][tid] = loss; red[1][tid] = skip; red[2][tid] = pd; red[3][tid] = nd;
  __syncthreads();
  for (int s = 128; s > 0; s >>= 1) {
    if (tid < s) {
      red[0][tid] += red[0][tid + s];
      red[1][tid] += red[1][tid + s];
      red[2][tid] += red[2][tid + s];
      red[3][tid] += red[3][tid + s];
    }
    __syncthreads();
  }
  if (tid == 0) {
    out[0] = red[0][0] / (float)NROW;                          // loss
    out[1] = red[1][0] / (float)NROW;                          // prec (skipped frac)
    out[2] = red[2][0] / ((float)NROW * 3.0f);                 // pos_d
    out[3] = red[3][0] / ((float)NROW * (float)(NROW - 4));    // neg_d
  }
}

extern "C" void kernel_launch(void* const* d_in, const int* in_sizes, int n_in,
                              void* d_out, int out_size, void* d_ws, size_t ws_size,
                              hipStream_t stream) {
  (void)in_sizes; (void)n_in; (void)out_size; (void)ws_size;
  const float* X = (const float*)d_in[0];   // [8192, 128] fp32
  const int*   T = (const int*)d_in[1];     // [8192] int32
  float* out = (float*)d_out;               // 4 floats

  float* ws     = (float*)d_ws;             // 8 * 8192 floats = 256 KB
  float* minpos = ws + 0 * NROW;
  float* maxneg = ws + 1 * NROW;
  float* posd   = ws + 2 * NROW;
  float* negd   = ws + 3 * NROW;
  float* pexp   = ws + 4 * NROW;
  float* nexp   = ws + 5 * NROW;
  float* pcnt   = ws + 6 * NROW;
  float* ncnt   = ws + 7 * NROW;

  lifted_pass1<<<NTILE, 256, 0, stream>>>(X, T, minpos, maxneg, posd, negd);
  lifted_pass2<<<NTILE, 256, 0, stream>>>(X, T, minpos, maxneg, pexp, nexp, pcnt, ncnt);
  lifted_finalize<<<1, 256, 0, stream>>>(pexp, nexp, pcnt, ncnt, posd, negd, out);
}